// Net_38560216384189
// MI455X (gfx1250) — compile-verified
//
#include <hip/hip_runtime.h>
#include <hip/hip_bf16.h>
#include <math.h>

typedef __attribute__((ext_vector_type(2))) float v2f;
typedef __attribute__((ext_vector_type(8))) float v8f;

#define F_IN 512
#define HC1  64     // 8 heads * 8 ch
#define H1   8
#define C1   8
#define C2   7
#define NEG_SLOPE 0.2f

__device__ __forceinline__ float lrelu(float x) {
    return x > 0.0f ? x : NEG_SLOPE * x;
}

// Monotonic float atomic max via signed/unsigned int ordering trick.
__device__ __forceinline__ void atomicMaxF(float* addr, float val) {
    if (val >= 0.0f)
        atomicMax((int*)addr, __float_as_int(val));
    else
        atomicMin((unsigned int*)addr, __float_as_uint(val));
}

// ---------------------------------------------------------------- utilities
__global__ void fill_kernel(float* __restrict__ p, float v, int n) {
    int i = blockIdx.x * blockDim.x + threadIdx.x;
    if (i < n) p[i] = v;
}

// ---------------------------------------------------------------- GEMM1 (WMMA fp32)
// H[N,64] = X[N,512] @ W1[512,64]; one wave computes a 16x16 tile,
// K-loop of 128 x V_WMMA_F32_16X16X4_F32. 8 waves / block; the 4 column
// tiles of the same 16 rows sit in the same block for A-row reuse.
__global__ __launch_bounds__(256) void gemm1_wmma(
        const float* __restrict__ X, const float* __restrict__ W1,
        float* __restrict__ H, int nrows) {
    const int lane   = threadIdx.x & 31;
    const int waveId = blockIdx.x * 8 + (threadIdx.x >> 5);
    const int rowTile = waveId >> 2;      // 4 col tiles (64 cols / 16)
    const int colTile = waveId & 3;
    const int rowBase = rowTile * 16;
    if (rowBase >= nrows) return;         // wave-uniform exit, EXEC stays full

    const int half = lane >> 4;           // 0: K={0,1}, 1: K={2,3}
    const int mrow = lane & 15;
    const int koff = half * 2;
    int xrow = rowBase + mrow;
    if (xrow >= nrows) xrow = nrows - 1;  // clamp keeps EXEC all-ones
    const int bcol = colTile * 16 + (lane & 15);

    const float* __restrict__ xp = X  + (size_t)xrow * F_IN + koff;
    const float* __restrict__ bp = W1 + (size_t)koff * HC1 + bcol;

    v8f c = {};
    for (int k = 0; k < F_IN; k += 4) {
        v2f a, b;
        a.x = xp[k];
        a.y = xp[k + 1];
        b.x = bp[(size_t)k * HC1];
        b.y = bp[(size_t)(k + 1) * HC1];
        c = __builtin_amdgcn_wmma_f32_16x16x4_f32(
                false, a, false, b, (short)0, c, false, false);
    }

    const int col = colTile * 16 + (lane & 15);
#pragma unroll
    for (int r = 0; r < 8; ++r) {
        int row = rowBase + r + 8 * half;
        if (row < nrows) H[(size_t)row * HC1 + col] = c[r];
    }
}

// ---------------------------------------------------------------- per-node attention logits (layer 1)
__global__ __launch_bounds__(256) void alpha1_kernel(
        const float* __restrict__ H, const float* __restrict__ a_src,
        const float* __restrict__ a_dst, float* __restrict__ as,
        float* __restrict__ ad, int n) {
    int i = blockIdx.x * blockDim.x + threadIdx.x;   // node*8 + head
    if (i >= n * H1) return;
    int node = i >> 3, head = i & 7;
    const float* hp = H + (size_t)node * HC1 + head * C1;
    const float* s  = a_src + head * C1;
    const float* d  = a_dst + head * C1;
    float vs = 0.0f, vd = 0.0f;
#pragma unroll
    for (int c = 0; c < C1; ++c) {
        float h = hp[c];
        vs += h * s[c];
        vd += h * d[c];
    }
    as[i] = vs;
    ad[i] = vd;
}

// ---------------------------------------------------------------- edge passes, layer 1 (8 heads)
__global__ __launch_bounds__(256) void edge_max1(
        const int* __restrict__ ei, int E, int n,
        const float* __restrict__ as, const float* __restrict__ ad,
        float* __restrict__ m) {
    int e = blockIdx.x * blockDim.x + threadIdx.x;
    if (e >= E + n) return;
    int s = (e < E) ? ei[e]     : (e - E);
    int d = (e < E) ? ei[E + e] : (e - E);
#pragma unroll
    for (int h = 0; h < H1; ++h) {
        float v = lrelu(as[s * H1 + h] + ad[d * H1 + h]);
        atomicMaxF(&m[d * H1 + h], v);
    }
}

__global__ __launch_bounds__(256) void edge_sum1(
        const int* __restrict__ ei, int E, int n,
        const float* __restrict__ as, const float* __restrict__ ad,
        const float* __restrict__ m, float* __restrict__ den) {
    int e = blockIdx.x * blockDim.x + threadIdx.x;
    if (e >= E + n) return;
    int s = (e < E) ? ei[e]     : (e - E);
    int d = (e < E) ? ei[E + e] : (e - E);
#pragma unroll
    for (int h = 0; h < H1; ++h) {
        float v = lrelu(as[s * H1 + h] + ad[d * H1 + h]);
        atomicAdd(&den[d * H1 + h], expf(v - m[d * H1 + h]));
    }
}

// one thread per (edge, head): compute alpha, scatter-add 8 channels
__global__ __launch_bounds__(256) void edge_agg1(
        const int* __restrict__ ei, int E, int n,
        const float* __restrict__ as, const float* __restrict__ ad,
        const float* __restrict__ m, const float* __restrict__ den,
        const float* __restrict__ H, float* __restrict__ out) {
    int t = blockIdx.x * blockDim.x + threadIdx.x;
    if (t >= (E + n) * H1) return;
    int e = t >> 3, h = t & 7;
    int s = (e < E) ? ei[e]     : (e - E);
    int d = (e < E) ? ei[E + e] : (e - E);
    float v     = lrelu(as[s * H1 + h] + ad[d * H1 + h]);
    float alpha = expf(v - m[d * H1 + h]) / (den[d * H1 + h] + 1e-16f);
    const float* hp = H + (size_t)s * HC1 + h * C1;
    float* op       = out + (size_t)d * HC1 + h * C1;
#pragma unroll
    for (int c = 0; c < C1; ++c)
        atomicAdd(&op[c], hp[c] * alpha);
}

// ---------------------------------------------------------------- layer-2 transform (fused bias + GEMM 64x7 + logits)
__global__ __launch_bounds__(256) void layer2_transform(
        const float* __restrict__ out1, const float* __restrict__ b1,
        const float* __restrict__ W2, const float* __restrict__ a_src2,
        const float* __restrict__ a_dst2, float* __restrict__ h2,
        float* __restrict__ as2, float* __restrict__ ad2, int n) {
    int i = blockIdx.x * blockDim.x + threadIdx.x;
    if (i >= n) return;
    float acc[C2];
#pragma unroll
    for (int c = 0; c < C2; ++c) acc[c] = 0.0f;
    const float* xp = out1 + (size_t)i * HC1;
#pragma unroll 8
    for (int k = 0; k < HC1; ++k) {
        float xv = xp[k] + b1[k];
#pragma unroll
        for (int c = 0; c < C2; ++c)
            acc[c] += xv * W2[k * C2 + c];
    }
    float vs = 0.0f, vd = 0.0f;
#pragma unroll
    for (int c = 0; c < C2; ++c) {
        h2[(size_t)i * C2 + c] = acc[c];
        vs += acc[c] * a_src2[c];
        vd += acc[c] * a_dst2[c];
    }
    as2[i] = vs;
    ad2[i] = vd;
}

// ---------------------------------------------------------------- edge passes, layer 2 (1 head)
__global__ __launch_bounds__(256) void edge_max2(
        const int* __restrict__ ei, int E, int n,
        const float* __restrict__ as, const float* __restrict__ ad,
        float* __restrict__ m) {
    int e = blockIdx.x * blockDim.x + threadIdx.x;
    if (e >= E + n) return;
    int s = (e < E) ? ei[e]     : (e - E);
    int d = (e < E) ? ei[E + e] : (e - E);
    atomicMaxF(&m[d], lrelu(as[s] + ad[d]));
}

__global__ __launch_bounds__(256) void edge_sum2(
        const int* __restrict__ ei, int E, int n,
        const float* __restrict__ as, const float* __restrict__ ad,
        const float* __restrict__ m, float* __restrict__ den) {
    int e = blockIdx.x * blockDim.x + threadIdx.x;
    if (e >= E + n) return;
    int s = (e < E) ? ei[e]     : (e - E);
    int d = (e < E) ? ei[E + e] : (e - E);
    float v = lrelu(as[s] + ad[d]);
    atomicAdd(&den[d], expf(v - m[d]));
}

__global__ __launch_bounds__(256) void edge_agg2(
        const int* __restrict__ ei, int E, int n,
        const float* __restrict__ as, const float* __restrict__ ad,
        const float* __restrict__ m, const float* __restrict__ den,
        const float* __restrict__ h2, float* __restrict__ out) {
    int e = blockIdx.x * blockDim.x + threadIdx.x;
    if (e >= E + n) return;
    int s = (e < E) ? ei[e]     : (e - E);
    int d = (e < E) ? ei[E + e] : (e - E);
    float v     = lrelu(as[s] + ad[d]);
    float alpha = expf(v - m[d]) / (den[d] + 1e-16f);
    const float* hp = h2 + (size_t)s * C2;
    float* op       = out + (size_t)d * C2;
#pragma unroll
    for (int c = 0; c < C2; ++c)
        atomicAdd(&op[c], hp[c] * alpha);
}

// ---------------------------------------------------------------- final bias + log_softmax
__global__ __launch_bounds__(256) void logsoftmax_kernel(
        const float* __restrict__ out2, const float* __restrict__ b2,
        float* __restrict__ y, int n) {
    int i = blockIdx.x * blockDim.x + threadIdx.x;
    if (i >= n) return;
    float v[C2];
    float mx = -INFINITY;
#pragma unroll
    for (int c = 0; c < C2; ++c) {
        v[c] = out2[(size_t)i * C2 + c] + b2[c];
        mx = fmaxf(mx, v[c]);
    }
    float s = 0.0f;
#pragma unroll
    for (int c = 0; c < C2; ++c) s += expf(v[c] - mx);
    float lse = mx + logf(s);
#pragma unroll
    for (int c = 0; c < C2; ++c)
        y[(size_t)i * C2 + c] = v[c] - lse;
}

// ---------------------------------------------------------------- launcher
extern "C" void kernel_launch(void* const* d_in, const int* in_sizes, int n_in,
                              void* d_out, int out_size, void* d_ws, size_t ws_size,
                              hipStream_t stream) {
    const float* x    = (const float*)d_in[0];
    const int*   ei   = (const int*)  d_in[1];
    const float* W1   = (const float*)d_in[2];
    const float* aS1  = (const float*)d_in[3];
    const float* aD1  = (const float*)d_in[4];
    const float* b1   = (const float*)d_in[5];
    const float* W2   = (const float*)d_in[6];
    const float* aS2  = (const float*)d_in[7];
    const float* aD2  = (const float*)d_in[8];
    const float* b2   = (const float*)d_in[9];

    const int N  = in_sizes[0] / F_IN;
    const int E  = in_sizes[1] / 2;
    const int ET = E + N;

    // workspace carve-up (floats)
    float* ws  = (float*)d_ws;
    size_t off = 0;
    float* h1   = ws + off; off += (size_t)N * HC1;
    float* as1  = ws + off; off += (size_t)N * H1;
    float* ad1  = ws + off; off += (size_t)N * H1;
    float* m1   = ws + off; off += (size_t)N * H1;
    float* den1 = ws + off; off += (size_t)N * H1;
    float* out1 = ws + off; off += (size_t)N * HC1;
    float* h2   = ws + off; off += (size_t)N * C2;
    float* as2  = ws + off; off += (size_t)N;
    float* ad2  = ws + off; off += (size_t)N;
    float* m2   = ws + off; off += (size_t)N;
    float* den2 = ws + off; off += (size_t)N;
    float* out2 = ws + off; off += (size_t)N * C2;

    const int TB = 256;
    auto blk = [](long long n, int tb) { return (unsigned)((n + tb - 1) / tb); };

    // ---- init accumulators (every call: graph replays must be deterministic)
    fill_kernel<<<blk((long long)N * H1,  TB), TB, 0, stream>>>(m1,  -INFINITY, N * H1);
    fill_kernel<<<blk((long long)N * H1,  TB), TB, 0, stream>>>(den1, 0.0f,     N * H1);
    fill_kernel<<<blk((long long)N * HC1, TB), TB, 0, stream>>>(out1, 0.0f,     N * HC1);
    fill_kernel<<<blk(N, TB), TB, 0, stream>>>(m2,  -INFINITY, N);
    fill_kernel<<<blk(N, TB), TB, 0, stream>>>(den2, 0.0f,     N);
    fill_kernel<<<blk((long long)N * C2, TB), TB, 0, stream>>>(out2, 0.0f, N * C2);

    // ---- layer 1
    {
        int rowTiles   = (N + 15) / 16;
        long long waves = (long long)rowTiles * 4;      // 4 col tiles
        gemm1_wmma<<<blk(waves, 8), 256, 0, stream>>>(x, W1, h1, N);
    }
    alpha1_kernel<<<blk((long long)N * H1, TB), TB, 0, stream>>>(h1, aS1, aD1, as1, ad1, N);
    edge_max1<<<blk(ET, TB), TB, 0, stream>>>(ei, E, N, as1, ad1, m1);
    edge_sum1<<<blk(ET, TB), TB, 0, stream>>>(ei, E, N, as1, ad1, m1, den1);
    edge_agg1<<<blk((long long)ET * H1, TB), TB, 0, stream>>>(ei, E, N, as1, ad1, m1, den1, h1, out1);

    // ---- layer 2
    layer2_transform<<<blk(N, TB), TB, 0, stream>>>(out1, b1, W2, aS2, aD2, h2, as2, ad2, N);
    edge_max2<<<blk(ET, TB), TB, 0, stream>>>(ei, E, N, as2, ad2, m2);
    edge_sum2<<<blk(ET, TB), TB, 0, stream>>>(ei, E, N, as2, ad2, m2, den2);
    edge_agg2<<<blk(ET, TB), TB, 0, stream>>>(ei, E, N, as2, ad2, m2, den2, h2, out2);

    // ---- output
    logsoftmax_kernel<<<blk(N, TB), TB, 0, stream>>>(out2, b2, (float*)d_out, N);
}